// CrossAttention_5007931867795
// MI455X (gfx1250) — compile-verified
//
#include <hip/hip_runtime.h>
#include <math.h>

// Problem sizes (fixed by the reference)
#define B_ 32
#define S_ 512
#define D_ 768

typedef __attribute__((ext_vector_type(2))) float v2f;
typedef __attribute__((ext_vector_type(4))) float v4f;
typedef __attribute__((ext_vector_type(8))) float v8f;
typedef int v4i_t __attribute__((__vector_size__(4 * sizeof(int))));

// GEMM modes
#define GM_PLAIN         0  // C = A @ B
#define GM_BIAS          1  // C = A @ B + aux0[row] + aux1[col]
#define GM_SOFT_ROWMAJOR 2  // A'[m,k] = exp(A[m,k]-aux0[k])*aux1[k]  (column softmax)
#define GM_SOFT_TRANS    3  // A'[m,k] = exp(A[k,m]-aux0[k])*aux1[k]  (row softmax, transposed)

#define KC    16            // K-chunk staged in LDS per iteration
#define LDA_P 20            // padded LDS stride for A rows (16B-aligned rows, banks spread)
#define LDB_P 136           // padded LDS stride for B rows (16B-aligned, half-waves disjoint banks)

// ---- CDNA5 async global->LDS copy (ASYNCcnt-tracked DMA), with fallback ----
#if __has_builtin(__builtin_amdgcn_global_load_async_to_lds_b128) && \
    __has_builtin(__builtin_amdgcn_s_wait_asynccnt)
#define HAVE_ASYNC 1
#define ASYNC_CP16(gsrc, ldst)                                   \
  __builtin_amdgcn_global_load_async_to_lds_b128(                \
      (__attribute__((address_space(1))) v4i_t*)(gsrc),          \
      (__attribute__((address_space(3))) v4i_t*)(ldst), 0, 0)
#define ASYNC_WAIT() __builtin_amdgcn_s_wait_asynccnt(0)
#else
#define HAVE_ASYNC 0
#define ASYNC_CP16(gsrc, ldst) ((void)0)
#define ASYNC_WAIT() ((void)0)
#endif

// ---------------------------------------------------------------------------
// LDS double-buffered fp32 WMMA GEMM.
//   Block: 256 threads (8 waves), tile 128x128. Wave: 32x64 = 2x4 WMMA tiles.
//   Untransformed operands stream global->LDS via async DMA (no VGPR staging);
//   exp-transformed A operands go global->regs->LDS with the softmax fused.
// ---------------------------------------------------------------------------
template <int MODE>
__global__ __launch_bounds__(256) void wmma_gemm_f32(
    const float* __restrict__ Aptr, int lda, long sA,
    const float* __restrict__ Bptr, int ldb, long sB,
    float* __restrict__ Cptr, int ldc, long sC,
    int K,
    const float* __restrict__ aux0,
    const float* __restrict__ aux1,
    int auxStride) {
  __shared__ float ldsA[2][128 * LDA_P];
  __shared__ float ldsB[2][KC * LDB_P];

  constexpr bool A_XFORM = (MODE == GM_SOFT_ROWMAJOR || MODE == GM_SOFT_TRANS);
  constexpr bool A_ASYNC = (HAVE_ASYNC != 0) && !A_XFORM;
  constexpr bool B_ASYNC = (HAVE_ASYNC != 0);

  const int tid  = threadIdx.x;
  const int wid  = tid >> 5;
  const int lane = tid & 31;
  const int half = lane >> 4;
  const int l16  = lane & 15;
  const int waveM = wid & 3;   // 4 waves along M
  const int waveN = wid >> 2;  // 2 waves along N
  const int b   = blockIdx.z;
  const int m0b = blockIdx.x * 128;   // block M origin
  const int n0b = blockIdx.y * 128;   // block N origin
  const int m0w = waveM * 32;         // wave M origin within block
  const int n0w = waveN * 64;         // wave N origin within block

  const float* A  = Aptr + (long)b * sA;
  const float* Bm = Bptr + (long)b * sB;
  float*       C  = Cptr + (long)b * sC;
  const float* a0p = aux0 ? (aux0 + (long)b * auxStride) : nullptr;
  const float* a1p = aux1 ? (aux1 + (long)b * auxStride) : nullptr;

  v8f acc[2][4];
#pragma unroll
  for (int i = 0; i < 2; ++i)
#pragma unroll
    for (int j = 0; j < 4; ++j)
#pragma unroll
      for (int r = 0; r < 8; ++r) acc[i][j][r] = 0.0f;

  // ---- staging registers (only used on non-async paths) ------------------
  v4f stA[2], stB[2];

  // Issue the A-operand fill for chunk k0 into buffer nbuf.
  auto issueA = [&](int k0, int nbuf) {
    if constexpr (A_ASYNC) {
      const int c4 = tid & 3, r = tid >> 2;
#pragma unroll
      for (int p = 0; p < 2; ++p)
        ASYNC_CP16(A + (long)(m0b + r + 64 * p) * lda + k0 + c4 * 4,
                   &ldsA[nbuf][(r + 64 * p) * LDA_P + c4 * 4]);
    } else if constexpr (MODE == GM_SOFT_TRANS) {
      (void)nbuf;
      const int m4 = tid & 31, kk = tid >> 5;  // 8 k-rows per pass, 2 passes
#pragma unroll
      for (int p = 0; p < 2; ++p) {
        const int krow = k0 + kk + 8 * p;
        v4f v = *(const v4f*)(A + (long)krow * lda + (m0b + m4 * 4));
        const float mx = a0p[krow], iv = a1p[krow];
#pragma unroll
        for (int e = 0; e < 4; ++e) v[e] = __expf(v[e] - mx) * iv;
        stA[p] = v;
      }
    } else {
      (void)nbuf;
      const int c4 = tid & 3, r = tid >> 2;    // 64 rows per pass, 2 passes
#pragma unroll
      for (int p = 0; p < 2; ++p) {
        v4f v = *(const v4f*)(A + (long)(m0b + r + 64 * p) * lda + k0 + c4 * 4);
        if (MODE == GM_SOFT_ROWMAJOR) {
          const v4f mx = *(const v4f*)(a0p + k0 + c4 * 4);
          const v4f iv = *(const v4f*)(a1p + k0 + c4 * 4);
#pragma unroll
          for (int e = 0; e < 4; ++e) v[e] = __expf(v[e] - mx[e]) * iv[e];
        }
        stA[p] = v;
      }
    }
  };

  auto issueB = [&](int k0, int nbuf) {
    const int n4 = tid & 31, kk = tid >> 5;
    if constexpr (B_ASYNC) {
#pragma unroll
      for (int p = 0; p < 2; ++p)
        ASYNC_CP16(Bm + (long)(k0 + kk + 8 * p) * ldb + n0b + n4 * 4,
                   &ldsB[nbuf][(kk + 8 * p) * LDB_P + n4 * 4]);
    } else {
      (void)nbuf;
#pragma unroll
      for (int p = 0; p < 2; ++p)
        stB[p] =
            *(const v4f*)(Bm + (long)(k0 + kk + 8 * p) * ldb + n0b + n4 * 4);
    }
  };

  // Register -> LDS stores for the non-async operands.
  auto regStore = [&](int nbuf) {
    if constexpr (!A_ASYNC) {
      if constexpr (MODE == GM_SOFT_TRANS) {
        const int m4 = tid & 31, kk = tid >> 5;
#pragma unroll
        for (int p = 0; p < 2; ++p)
#pragma unroll
          for (int e = 0; e < 4; ++e)
            ldsA[nbuf][(m4 * 4 + e) * LDA_P + kk + 8 * p] = stA[p][e];
      } else {
        const int c4 = tid & 3, r = tid >> 2;
#pragma unroll
        for (int p = 0; p < 2; ++p)
          *(v4f*)&ldsA[nbuf][(r + 64 * p) * LDA_P + c4 * 4] = stA[p];
      }
    }
    if constexpr (!B_ASYNC) {
      const int n4 = tid & 31, kk = tid >> 5;
#pragma unroll
      for (int p = 0; p < 2; ++p)
        *(v4f*)&ldsB[nbuf][(kk + 8 * p) * LDB_P + n4 * 4] = stB[p];
    }
  };

  // 32 WMMAs on one LDS-resident chunk.
  auto compute = [&](int buf) {
#pragma unroll
    for (int kk = 0; kk < KC; kk += 4) {
      const int ka = kk + 2 * half;
      v2f afrag[2];
#pragma unroll
      for (int i = 0; i < 2; ++i)
        afrag[i] = *(const v2f*)&ldsA[buf][(m0w + 16 * i + l16) * LDA_P + ka];
      v2f bfrag[4];
#pragma unroll
      for (int j = 0; j < 4; ++j) {
        const int n = n0w + 16 * j + l16;
        bfrag[j].x = ldsB[buf][ka * LDB_P + n];
        bfrag[j].y = ldsB[buf][(ka + 1) * LDB_P + n];
      }
#pragma unroll
      for (int i = 0; i < 2; ++i)
#pragma unroll
        for (int j = 0; j < 4; ++j)
          acc[i][j] = __builtin_amdgcn_wmma_f32_16x16x4_f32(
              false, afrag[i], false, bfrag[j], (short)0, acc[i][j], false,
              false);
    }
  };

  // ---- pipelined main loop ----------------------------------------------
  issueA(0, 0);
  issueB(0, 0);
  regStore(0);
  ASYNC_WAIT();
  __syncthreads();

  int buf = 0;
  for (int kc = 0; kc < K; kc += KC) {
    const bool more = (kc + KC) < K;
    if (more) {
      issueA(kc + KC, buf ^ 1);  // async variants land in back buffer now
      issueB(kc + KC, buf ^ 1);
    }
    if (kc + 2 * KC < K) {  // L2 prefetch two chunks ahead (global_prefetch_b8)
      const int kn = kc + 2 * KC;
      if (MODE == GM_SOFT_TRANS) {
        const int m4 = tid & 31, kk = tid >> 5;
        __builtin_prefetch(A + (long)(kn + kk) * lda + (m0b + m4 * 4), 0, 1);
        __builtin_prefetch(A + (long)(kn + kk + 8) * lda + (m0b + m4 * 4), 0, 1);
      } else {
        const int c4 = tid & 3, r = tid >> 2;
        __builtin_prefetch(A + (long)(m0b + r) * lda + kn + c4 * 4, 0, 1);
        __builtin_prefetch(A + (long)(m0b + r + 64) * lda + kn + c4 * 4, 0, 1);
      }
      const int n4 = tid & 31, kb = tid >> 5;
      __builtin_prefetch(Bm + (long)(kn + kb) * ldb + n0b + n4 * 4, 0, 1);
      __builtin_prefetch(Bm + (long)(kn + kb + 8) * ldb + n0b + n4 * 4, 0, 1);
    }
    compute(buf);
    if (more) {
      regStore(buf ^ 1);
      ASYNC_WAIT();
      __syncthreads();
      buf ^= 1;
    }
  }

  // ---- epilogue. C/D layout: VGPR r holds row (r + 8*half), col l16. -----
#pragma unroll
  for (int i = 0; i < 2; ++i) {
    const int mb = m0b + m0w + 16 * i + 8 * half;
#pragma unroll
    for (int j = 0; j < 4; ++j) {
      const int n = n0b + n0w + 16 * j + l16;
      float cb = 0.f;
      if (MODE == GM_BIAS) cb = a1p[n];
#pragma unroll
      for (int r = 0; r < 8; ++r) {
        float v = acc[i][j][r];
        if (MODE == GM_BIAS) v += a0p[mb + r] + cb;
        C[(long)(mb + r) * ldc + n] = v;
      }
    }
  }
}

// ---------------------------------------------------------------------------
// bias1[row] = dot(x1[row,:], W1); bias2[row] = dot(x2[row,:], W2)
// ---------------------------------------------------------------------------
__global__ __launch_bounds__(256) void bias_kernel(
    const float* __restrict__ x1, const float* __restrict__ x2,
    const float* __restrict__ W1, const float* __restrict__ W2,
    float* __restrict__ bias1, float* __restrict__ bias2) {
  const int lane = threadIdx.x & 31;
  const int row  = blockIdx.x * 8 + (threadIdx.x >> 5);
  const float* p1 = x1 + (long)row * D_;
  const float* p2 = x2 + (long)row * D_;
  float s1 = 0.f, s2 = 0.f;
#pragma unroll
  for (int i = 0; i < D_ / 32; ++i) {
    const int c = lane + 32 * i;
    s1 += p1[c] * W1[c];
    s2 += p2[c] * W2[c];
  }
#pragma unroll
  for (int off = 16; off > 0; off >>= 1) {
    s1 += __shfl_xor(s1, off, 32);
    s2 += __shfl_xor(s2, off, 32);
  }
  if (lane == 0) {
    bias1[row] = s1;
    bias2[row] = s2;
  }
}

// ---------------------------------------------------------------------------
// Row softmax stats (axis=2): one wave per row of the SxS score matrix.
// ---------------------------------------------------------------------------
__global__ __launch_bounds__(256) void row_stats_kernel(
    const float* __restrict__ Sc, float* __restrict__ rmax,
    float* __restrict__ rinv) {
  const int lane = threadIdx.x & 31;
  const int row  = blockIdx.x * 8 + (threadIdx.x >> 5);
  const float* p = Sc + (long)row * S_;
  float vals[S_ / 32];
  float m = -3.0e38f;
#pragma unroll
  for (int i = 0; i < S_ / 32; ++i) {
    vals[i] = p[lane + 32 * i];
    m = fmaxf(m, vals[i]);
  }
#pragma unroll
  for (int off = 16; off > 0; off >>= 1) m = fmaxf(m, __shfl_xor(m, off, 32));
  float s = 0.f;
#pragma unroll
  for (int i = 0; i < S_ / 32; ++i) s += __expf(vals[i] - m);
#pragma unroll
  for (int off = 16; off > 0; off >>= 1) s += __shfl_xor(s, off, 32);
  if (lane == 0) {
    rmax[row] = m;
    rinv[row] = 1.0f / s;
  }
}

// ---------------------------------------------------------------------------
// Column softmax stats (axis=1): one thread per column, coalesced row sweep.
// ---------------------------------------------------------------------------
__global__ __launch_bounds__(256) void col_stats_kernel(
    const float* __restrict__ Sc, float* __restrict__ cmax,
    float* __restrict__ cinv) {
  const int b = blockIdx.y;
  const int t = blockIdx.x * 256 + threadIdx.x;
  const float* p = Sc + (long)b * S_ * S_ + t;
  float m = -3.0e38f;
  for (int r = 0; r < S_; ++r) m = fmaxf(m, p[(long)r * S_]);
  float s = 0.f;
  for (int r = 0; r < S_; ++r) s += __expf(p[(long)r * S_] - m);
  cmax[b * S_ + t] = m;
  cinv[b * S_ + t] = 1.0f / s;
}

// ---------------------------------------------------------------------------
extern "C" void kernel_launch(void* const* d_in, const int* in_sizes, int n_in,
                              void* d_out, int out_size, void* d_ws,
                              size_t ws_size, hipStream_t stream) {
  (void)in_sizes; (void)n_in; (void)out_size; (void)ws_size;
  const float* x1 = (const float*)d_in[0];
  const float* x2 = (const float*)d_in[1];
  const float* Wc = (const float*)d_in[2];
  const float* W1 = (const float*)d_in[3];
  const float* W2 = (const float*)d_in[4];

  float* fx1 = (float*)d_out;               // [B,S,D]
  float* fx2 = fx1 + (size_t)B_ * S_ * D_;  // [B,S,D]

  float* ws    = (float*)d_ws;
  float* x1w   = ws;  ws += (size_t)B_ * S_ * D_;  // 48 MB
  float* Sc    = ws;  ws += (size_t)B_ * S_ * S_;  // 32 MB
  float* bias1 = ws;  ws += B_ * S_;
  float* bias2 = ws;  ws += B_ * S_;
  float* rmax  = ws;  ws += B_ * S_;
  float* rinv  = ws;  ws += B_ * S_;
  float* cmax  = ws;  ws += B_ * S_;
  float* cinv  = ws;  ws += B_ * S_;

  // 1) additive bias terms
  bias_kernel<<<(B_ * S_) / 8, 256, 0, stream>>>(x1, x2, W1, W2, bias1, bias2);

  // 2) x1w = x1 @ Wc   (flattened: M = B*S, N = K = D)
  wmma_gemm_f32<GM_PLAIN>
      <<<dim3((B_ * S_) / 128, D_ / 128, 1), 256, 0, stream>>>(
          x1, D_, 0, Wc, D_, 0, x1w, D_, 0, D_, nullptr, nullptr, 0);

  // 3) Sc = x1w @ reshape(x2 -> DxS) + bias1[s] + bias2[t]   (per batch)
  wmma_gemm_f32<GM_BIAS>
      <<<dim3(S_ / 128, S_ / 128, B_), 256, 0, stream>>>(
          x1w, D_, (long)S_ * D_, x2, S_, (long)S_ * D_, Sc, S_,
          (long)S_ * S_, D_, bias1, bias2, S_);

  // 4) softmax statistics for both axes
  row_stats_kernel<<<(B_ * S_) / 8, 256, 0, stream>>>(Sc, rmax, rinv);
  col_stats_kernel<<<dim3(S_ / 256, B_), 256, 0, stream>>>(Sc, cmax, cinv);

  // 5) f_x1 = colsoftmax(Sc) @ x2   (softmax fused into LDS fill)
  wmma_gemm_f32<GM_SOFT_ROWMAJOR>
      <<<dim3(S_ / 128, D_ / 128, B_), 256, 0, stream>>>(
          Sc, S_, (long)S_ * S_, x2, D_, (long)S_ * D_, fx1, D_,
          (long)S_ * D_, S_, cmax, cinv, S_);

  // 6) f_x2 = rowsoftmax(Sc)^T @ x1 (transpose folded into LDS scatter)
  wmma_gemm_f32<GM_SOFT_TRANS>
      <<<dim3(S_ / 128, D_ / 128, B_), 256, 0, stream>>>(
          Sc, S_, (long)S_ * S_, x1, D_, (long)S_ * D_, fx2, D_,
          (long)S_ * D_, S_, rmax, rinv, S_);
}